// graphcl_82248623719027
// MI455X (gfx1250) — compile-verified
//
#include <hip/hip_runtime.h>
#include <hip/hip_bf16.h>

#define DIM 300
#define BGRAPHS 8192
#define CHUNKS 75          // float4 chunks per 300-float row
#define NODES_PER_BLOCK 64
#define MTILES 4           // 16-row M-tiles per wave in the GEMM

typedef __attribute__((ext_vector_type(2))) float v2f;
typedef __attribute__((ext_vector_type(8))) float v8f;

// ---------------- zero init of workspace region ----------------
__global__ void zero_kernel(float* __restrict__ p, long long n) {
    long long i = (long long)blockIdx.x * blockDim.x + threadIdx.x;
    long long stride = (long long)gridDim.x * blockDim.x;
    for (; i < n; i += stride) p[i] = 0.0f;
}

// ---------------- segment max + counts ----------------
__global__ void segstats_kernel(const float* __restrict__ node_imp,
                                const int* __restrict__ batch,
                                float* __restrict__ seg_max,   // bitwise int max (imp >= 0)
                                float* __restrict__ counts,
                                int N) {
    int i = blockIdx.x * blockDim.x + threadIdx.x;
    if (i >= N) return;
    int seg = batch[i];
    float imp = node_imp[i];
    atomicMax((int*)&seg_max[seg], __float_as_int(imp));
    atomicAdd(&counts[seg], 1.0f);
}

// ---------------- scaled segment sum, exploiting sorted batch ----------------
__global__ void scatter_sum_kernel(const float* __restrict__ x,
                                   const float* __restrict__ node_imp,
                                   const int* __restrict__ batch,
                                   const float* __restrict__ seg_max,
                                   float* __restrict__ sums,
                                   int N) {
    int t = threadIdx.x;                 // blockDim = 96, active t < 75
    bool active = t < CHUNKS;
    long long start = (long long)blockIdx.x * NODES_PER_BLOCK;
    long long end   = start + NODES_PER_BLOCK;
    if (end > N) end = N;
    if (start >= end) return;

    float4 acc = make_float4(0.f, 0.f, 0.f, 0.f);
    int curseg = batch[start];           // uniform across block

    for (long long i = start; i < end; ++i) {
        int seg = batch[i];              // uniform
        if (seg != curseg) {             // uniform branch: flush run
            if (active) {
                float* base = sums + (long long)curseg * DIM + t * 4;
                atomicAdd(base + 0, acc.x);
                atomicAdd(base + 1, acc.y);
                atomicAdd(base + 2, acc.z);
                atomicAdd(base + 3, acc.w);
            }
            acc = make_float4(0.f, 0.f, 0.f, 0.f);
            curseg = seg;
        }
        float scale = node_imp[i] / (seg_max[seg] * 10.0f) + 0.9f;
        if (active) {
            const float4 v = *(const float4*)(x + (long long)i * DIM + t * 4);
            acc.x += v.x * scale;
            acc.y += v.y * scale;
            acc.z += v.z * scale;
            acc.w += v.w * scale;
        }
    }
    if (active) {
        float* base = sums + (long long)curseg * DIM + t * 4;
        atomicAdd(base + 0, acc.x);
        atomicAdd(base + 1, acc.y);
        atomicAdd(base + 2, acc.z);
        atomicAdd(base + 3, acc.w);
    }
}

// ---------------- mean divide (in place) ----------------
__global__ void pool_div_kernel(float* __restrict__ sums,
                                const float* __restrict__ counts) {
    long long i = (long long)blockIdx.x * blockDim.x + threadIdx.x;
    long long total = (long long)BGRAPHS * DIM;
    if (i >= total) return;
    float cnt = counts[i / DIM];
    sums[i] = sums[i] / fmaxf(cnt, 1.0f);
}

// ---------------- WMMA fp32 GEMM: out = [relu](A @ W^T + bias) ----------------
// A: (M x K=300) row-major; W: (300 x 300) row-major (so B[k][n] = W[n*K+k]).
// One wave per 64x16 output macro-tile (4 stacked 16x16 WMMA tiles sharing B);
// V_WMMA_F32_16X16X4_F32, K stepped by 4. Full N-tiles take a guard-free path.
template <bool RELU>
__global__ void gemm_nt_bias_kernel(const float* __restrict__ A,
                                    const float* __restrict__ W,
                                    const float* __restrict__ bias,
                                    float* __restrict__ out) {
    const int K = DIM, NCOLS = DIM;
    int m0 = blockIdx.x * (16 * MTILES);
    int n0 = blockIdx.y * 16;
    int lane = threadIdx.x;        // 0..31
    int half = lane >> 4;          // 0: K0/K1, 1: K2/K3
    int l15  = lane & 15;
    int koff = half * 2;

    int ncol = n0 + l15;           // B/D: N = lane & 15
    bool fullTile = (n0 + 16 <= NCOLS);     // uniform across wave
    bool nOK = ncol < NCOLS;
    int ncolc = nOK ? ncol : (NCOLS - 1);   // clamped address, zero-selected value

    const float* aBase = A + (long long)(m0 + l15) * K + koff;   // tile j adds j*16*K
    const float* bBase = W + (long long)ncolc * K + koff;

    v8f c[MTILES];
    #pragma unroll
    for (int j = 0; j < MTILES; ++j) c[j] = (v8f){};

    if (fullTile) {
        for (int k = 0; k < K; k += 4) {
            v2f b;
            b.x = bBase[k];
            b.y = bBase[k + 1];
            #pragma unroll
            for (int j = 0; j < MTILES; ++j) {
                v2f a;
                a.x = aBase[(long long)j * 16 * K + k];
                a.y = aBase[(long long)j * 16 * K + k + 1];
                c[j] = __builtin_amdgcn_wmma_f32_16x16x4_f32(
                        false, a, false, b, (short)0, c[j], false, false);
            }
        }
    } else {
        for (int k = 0; k < K; k += 4) {
            float bx = bBase[k];
            float by = bBase[k + 1];
            v2f b;
            b.x = nOK ? bx : 0.0f;   // keep EXEC full at the WMMA (ISA requirement)
            b.y = nOK ? by : 0.0f;
            #pragma unroll
            for (int j = 0; j < MTILES; ++j) {
                v2f a;
                a.x = aBase[(long long)j * 16 * K + k];
                a.y = aBase[(long long)j * 16 * K + k + 1];
                c[j] = __builtin_amdgcn_wmma_f32_16x16x4_f32(
                        false, a, false, b, (short)0, c[j], false, false);
            }
        }
    }

    if (nOK) {
        float bv = bias[ncol];
        #pragma unroll
        for (int j = 0; j < MTILES; ++j) {
            int rowbase = m0 + j * 16 + half * 8;  // lanes 0-15: M=r, 16-31: M=r+8
            #pragma unroll
            for (int r = 0; r < 8; ++r) {
                float v = c[j][r] + bv;
                if (RELU) v = fmaxf(v, 0.0f);
                out[(long long)(rowbase + r) * NCOLS + ncol] = v;
            }
        }
    }
}

extern "C" void kernel_launch(void* const* d_in, const int* in_sizes, int n_in,
                              void* d_out, int out_size, void* d_ws, size_t ws_size,
                              hipStream_t stream) {
    const float* x        = (const float*)d_in[0];
    const float* node_imp = (const float*)d_in[1];
    const float* W1       = (const float*)d_in[2];
    const float* b1       = (const float*)d_in[3];
    const float* W2       = (const float*)d_in[4];
    const float* b2       = (const float*)d_in[5];
    const int*   batch    = (const int*)d_in[6];
    float* out = (float*)d_out;

    int N = in_sizes[0] / DIM;

    // workspace layout (floats)
    float* ws       = (float*)d_ws;
    float* seg_max  = ws;                          // BGRAPHS
    float* counts   = seg_max + BGRAPHS;           // BGRAPHS
    float* sums     = counts + BGRAPHS;            // BGRAPHS*DIM (pooled in place)
    float* h        = sums + (long long)BGRAPHS * DIM;  // BGRAPHS*DIM

    long long zeroN = 2LL * BGRAPHS + (long long)BGRAPHS * DIM;
    zero_kernel<<<2048, 256, 0, stream>>>(ws, zeroN);

    int blocksStats = (N + 255) / 256;
    segstats_kernel<<<blocksStats, 256, 0, stream>>>(node_imp, batch, seg_max, counts, N);

    int blocksScatter = (N + NODES_PER_BLOCK - 1) / NODES_PER_BLOCK;
    scatter_sum_kernel<<<blocksScatter, 96, 0, stream>>>(x, node_imp, batch, seg_max, sums, N);

    long long poolN = (long long)BGRAPHS * DIM;
    pool_div_kernel<<<(int)((poolN + 255) / 256), 256, 0, stream>>>(sums, counts);

    dim3 gemmGrid(BGRAPHS / (16 * MTILES), (DIM + 15) / 16);  // 128 x 19 macro-tiles
    gemm_nt_bias_kernel<true ><<<gemmGrid, 32, 0, stream>>>(sums, W1, b1, h);
    gemm_nt_bias_kernel<false><<<gemmGrid, 32, 0, stream>>>(h,    W2, b2, out);
}